// HungarianMatcher_64287070486644
// MI455X (gfx1250) — compile-verified
//
#include <hip/hip_runtime.h>
#include <hip/hip_bf16.h>

// Problem constants (from reference setup_inputs)
#define BB   16    // batches
#define NQ   900   // queries per batch
#define TT   64    // targets per batch
#define LL   96    // feature length (4 bbox + 1 unused + 91 classes)
#define CC   91    // classes
#define BIGF 1e9f

typedef __attribute__((ext_vector_type(2))) float v2f;
typedef __attribute__((ext_vector_type(8))) float v8f;

// ---------------------------------------------------------------------------
// Phase A: cost matrix, stored transposed costT[b][t][n] (t<64 rows, n<900).
// Class cost as a one-hot GEMM via V_WMMA_F32_16X16X4_F32 (exact f32 math):
//   D[16x16] += A[16x4] (prob tile) x B[4x16] (one-hot of target classes)
// One wave per (batch, 16-query tile); 4 N-tiles x 23 K-steps of WMMA.
// ---------------------------------------------------------------------------
__global__ __launch_bounds__(32)
void cost_kernel(const float* __restrict__ outputs,
                 const float* __restrict__ targets,
                 float* __restrict__ costT) {
  const int b     = blockIdx.y;
  const int mbase = blockIdx.x * 16;
  const int lane  = threadIdx.x;        // 0..31, wave32
  const int lm    = lane & 15;
  const int half  = lane >> 4;          // 0: K{0,1}, 1: K{2,3} per ISA A/B layout

  __shared__ float sP[16][92];          // prob tile, zero-padded to K=92
  __shared__ float sTB[64][4];          // target boxes (xyxy)
  __shared__ int   sTC[64];             // target classes
  __shared__ float sOB[16][4];          // query boxes (xywh)

  // Stage probability tile into LDS (rows clamped beyond N=900; unused anyway).
  for (int idx = lane; idx < 16 * 92; idx += 32) {
    int r = idx / 92, c = idx - r * 92;
    int row = mbase + r; if (row >= NQ) row = NQ - 1;
    sP[r][c] = (c < CC) ? outputs[((size_t)b * NQ + row) * LL + 5 + c] : 0.0f;
  }
  for (int idx = lane; idx < TT; idx += 32) {
    const float* tp = targets + ((size_t)b * TT + idx) * 5;
    sTB[idx][0] = tp[0]; sTB[idx][1] = tp[1];
    sTB[idx][2] = tp[2]; sTB[idx][3] = tp[3];
    sTC[idx] = (int)tp[4];
  }
  for (int idx = lane; idx < 64; idx += 32) {
    int r = idx >> 2, k = idx & 3;
    int row = mbase + r; if (row >= NQ) row = NQ - 1;
    sOB[r][k] = outputs[((size_t)b * NQ + row) * LL + k];
  }
  __syncthreads();

#pragma unroll
  for (int nt = 0; nt < 4; ++nt) {
    const int t   = nt * 16 + lm;       // B/D column owned by this lane
    const int cls = sTC[t];

    v8f acc = {0.f, 0.f, 0.f, 0.f, 0.f, 0.f, 0.f, 0.f};
#pragma unroll
    for (int kk = 0; kk < 23; ++kk) {
      const int c0 = kk * 4 + half * 2; // K indices for this lane's 2 VGPRs
      v2f a = *(const v2f*)&sP[lm][c0]; // A: 16x4 f32 fragment
      v2f bb;                            // B: 4x16 one-hot fragment (registers)
      bb.x = (cls == c0)     ? 1.0f : 0.0f;
      bb.y = (cls == c0 + 1) ? 1.0f : 0.0f;
      acc = __builtin_amdgcn_wmma_f32_16x16x4_f32(
          /*neg_a=*/false, a, /*neg_b=*/false, bb,
          /*c_mod=*/(short)0, acc, /*reuse_a=*/false, /*reuse_b=*/false);
    }
    // acc[r] = out_prob[mbase + r + 8*half, cls[t]]  (C/D layout per ISA)

    const float tbx = sTB[t][0], tby = sTB[t][1];
    const float tbz = sTB[t][2], tbw = sTB[t][3];
    const float a2  = (tbz - tbx) * (tbw - tby);
#pragma unroll
    for (int r = 0; r < 8; ++r) {
      const int ml = r + half * 8;
      const float obx = sOB[ml][0], oby = sOB[ml][1];
      const float obz = sOB[ml][2], obw = sOB[ml][3];
      // L1 between xywh query box and xyxy target box (faithful to reference)
      float cb = fabsf(obx - tbx) + fabsf(oby - tby) +
                 fabsf(obz - tbz) + fabsf(obw - tbw);
      // GIoU on converted query box
      float ox0 = obx - obz * 0.5f, oy0 = oby - obw * 0.5f;
      float ox1 = obx + obz * 0.5f, oy1 = oby + obw * 0.5f;
      float a1  = (ox1 - ox0) * (oy1 - oy0);
      float ltx = fmaxf(ox0, tbx), lty = fmaxf(oy0, tby);
      float rbx = fminf(ox1, tbz), rby = fminf(oy1, tbw);
      float iw  = fmaxf(rbx - ltx, 0.f), ih = fmaxf(rby - lty, 0.f);
      float inter = iw * ih;
      float uni   = a1 + a2 - inter;
      float iou   = inter / uni;
      float lex = fminf(ox0, tbx), ley = fminf(oy0, tby);
      float rex = fmaxf(ox1, tbz), rey = fmaxf(oy1, tbw);
      float ew  = fmaxf(rex - lex, 0.f), eh = fmaxf(rey - ley, 0.f);
      float ae  = ew * eh;
      float giou  = iou - (ae - uni) / ae;
      float total = 5.0f * cb - 2.0f * giou - acc[r];
      int n = mbase + ml;
      if (n < NQ) costT[((size_t)b * TT + t) * NQ + n] = total;
    }
  }
}

// ---------------------------------------------------------------------------
// Phase B: Jonker-Volgenant shortest augmenting path on costT[b] ([T x N]).
// One block per batch; 1024 threads; column j owned by thread j (N+1=901).
// Serial control on thread 0, 901-wide scans + LDS argmin tree in parallel.
// ---------------------------------------------------------------------------
__global__ __launch_bounds__(1024)
void hungarian_kernel(const float* __restrict__ costT, int* __restrict__ out) {
  const int b   = blockIdx.x;
  const int tid = threadIdx.x;
  const float* Cb = costT + (size_t)b * TT * NQ;

  __shared__ float v_[NQ + 1];
  __shared__ float minv[NQ + 1];
  __shared__ float u_[TT + 2];
  __shared__ int   used[NQ + 1];
  __shared__ int   way[NQ + 1];
  __shared__ int   p_[NQ + 1];
  __shared__ float redv[1024];
  __shared__ int   redi[1024];
  __shared__ int   q_[TT];

  if (tid <= NQ) { v_[tid] = 0.f; p_[tid] = 0; }
  if (tid <= TT) u_[tid] = 0.f;
  __syncthreads();

  for (int i = 1; i <= TT; ++i) {
    if (tid == 0) p_[0] = i;
    if (tid <= NQ) { minv[tid] = BIGF; used[tid] = 0; way[tid] = 0; }
    __syncthreads();

    int j0 = 0;
    while (true) {
      const int i0 = p_[j0];
      if (tid == 0) used[j0] = 1;
      __syncthreads();

      const float ui0 = u_[i0];
      float lb = BIGF; int lj = tid;
      if (tid <= NQ && !used[tid]) {
        float cj  = (tid == 0) ? 0.f : Cb[(size_t)(i0 - 1) * NQ + (tid - 1)];
        float cur = cj - ui0 - v_[tid];
        if (cur < minv[tid]) { minv[tid] = cur; way[tid] = j0; }
        lb = minv[tid];
      }
      redv[tid] = lb; redi[tid] = lj;
      for (int s = 512; s > 0; s >>= 1) {
        __syncthreads();
        if (tid < s) {
          float ov = redv[tid + s]; int oj = redi[tid + s];
          if (ov < redv[tid] || (ov == redv[tid] && oj < redi[tid])) {
            redv[tid] = ov; redi[tid] = oj;   // argmin, first-index tie-break
          }
        }
      }
      __syncthreads();
      const float delta = redv[0];
      const int   j1    = redi[0];

      if (tid <= NQ) {
        if (used[tid]) {
          u_[p_[tid]] += delta;   // distinct rows per used column: race-free
          v_[tid]     -= delta;
        } else {
          minv[tid]   -= delta;
        }
      }
      __syncthreads();

      j0 = j1;
      if (p_[j0] == 0) break;     // uniform across block
    }

    if (tid == 0) {               // augment along the alternating path
      int jj = j0;
      while (jj != 0) {
        int jp = way[jj];
        p_[jj] = p_[jp];
        jj = jp;
      }
    }
    __syncthreads();
  }

  // Invert matching: q[row-1] = column-1
  if (tid >= 1 && tid <= NQ) {
    int r = p_[tid];
    if (r >= 1) q_[r - 1] = tid - 1;
  }
  __syncthreads();

  // Sort pairs by query index (rank = count of smaller q; q values distinct)
  if (tid < TT) {
    int myq = q_[tid];
    int rank = 0;
    for (int s = 0; s < TT; ++s) rank += (q_[s] < myq) ? 1 : 0;
    out[b * TT + rank]           = myq;  // i_idx (query)
    out[BB * TT + b * TT + rank] = tid;  // j_idx (target)
  }
}

// ---------------------------------------------------------------------------
extern "C" void kernel_launch(void* const* d_in, const int* in_sizes, int n_in,
                              void* d_out, int out_size, void* d_ws, size_t ws_size,
                              hipStream_t stream) {
  const float* outputs = (const float*)d_in[0];   // [16,900,96] f32
  const float* targets = (const float*)d_in[1];   // [16,64,5]  f32
  float* costT = (float*)d_ws;                    // [16,64,900] f32 (3.7 MB)
  int* out = (int*)d_out;                         // i_idx[16,64] ++ j_idx[16,64]

  dim3 g1((NQ + 15) / 16, BB);                    // 57 x 16 blocks, 1 wave each
  cost_kernel<<<g1, 32, 0, stream>>>(outputs, targets, costT);
  hungarian_kernel<<<BB, 1024, 0, stream>>>(costT, out);
}